// Interp_SocialLSTM_77721728189067
// MI455X (gfx1250) — compile-verified
//
#include <hip/hip_runtime.h>
#include <math.h>

#define NN 384
#define RR 128
#define KK 4
#define GG 5
#define OO 2

typedef float v2f __attribute__((ext_vector_type(2)));
typedef float v8f __attribute__((ext_vector_type(8)));

__device__ __forceinline__ float sigmoidf_(float x) { return 1.0f / (1.0f + expf(-x)); }

// ---------------------------------------------------------------------------
// K0: Csum[k] = sum_r (W_mode[r][k] + b_mode[r])   (tiny, 1 block)
// ---------------------------------------------------------------------------
__global__ void k0_csum(const float* __restrict__ W_mode,
                        const float* __restrict__ b_mode,
                        float* __restrict__ csum) {
    __shared__ float red[RR * KK];
    int r = threadIdx.x;  // 128 threads
    float bm = b_mode[r];
    for (int k = 0; k < KK; ++k) red[r * KK + k] = W_mode[r * KK + k] + bm;
    __syncthreads();
    if (r < KK) {
        float s = 0.f;
        for (int i = 0; i < RR; ++i) s += red[i * KK + r];
        csum[r] = s;
    }
}

// ---------------------------------------------------------------------------
// K1: per (n,i): head scores s_k, argmax k*, masked sv = sum_r loc[:,k*] + C[k*]
// grid (NN, 3), block 128  (thread = one neighbor i)
// ---------------------------------------------------------------------------
__global__ void k1_pass1(const float* __restrict__ nodes,
                         const int* __restrict__ visible,
                         const float* __restrict__ h,
                         const float* __restrict__ W_loc,
                         const float* __restrict__ b_loc,
                         const float* __restrict__ W_score,
                         const float* __restrict__ csum,
                         float* __restrict__ sv,
                         int* __restrict__ kstar) {
    __shared__ float wl[512 * 6];
    __shared__ float bl[512];
    __shared__ float hw[RR];
    __shared__ float nd[6];
    __shared__ float cs[KK];
    int n = blockIdx.x;
    int tid = threadIdx.x;
    for (int j = tid; j < 512 * 6; j += 128) wl[j] = W_loc[j];
    for (int j = tid; j < 512; j += 128) bl[j] = b_loc[j];
    hw[tid] = h[n * RR + tid] * W_score[tid];
    if (tid < 6) nd[tid] = nodes[n * 6 + tid];
    if (tid < KK) cs[tid] = csum[tid];
    __syncthreads();

    int i = blockIdx.y * 128 + tid;
    float r0 = nd[0] - nodes[i * 6 + 0];
    float r1 = nd[1] - nodes[i * 6 + 1];
    float r2 = nd[2] - nodes[i * 6 + 2];
    float r3 = nd[3] - nodes[i * 6 + 3];
    float r4 = nd[4] - nodes[i * 6 + 4];
    float r5 = nd[5] - nodes[i * 6 + 5];

    float s0 = 0.f, s1 = 0.f, s2 = 0.f, s3 = 0.f;
    float m0 = 0.f, m1 = 0.f, m2 = 0.f, m3 = 0.f;
    for (int r = 0; r < RR; ++r) {
        float hwr = hw[r];
        const float* w = &wl[r * 4 * 6];
        float d, l;
        d = bl[r * 4 + 0] + w[0] * r0 + w[1] * r1 + w[2] * r2 + w[3] * r3 + w[4] * r4 + w[5] * r5;
        l = fmaxf(d, 0.f); s0 += l * hwr; m0 += l;
        d = bl[r * 4 + 1] + w[6] * r0 + w[7] * r1 + w[8] * r2 + w[9] * r3 + w[10] * r4 + w[11] * r5;
        l = fmaxf(d, 0.f); s1 += l * hwr; m1 += l;
        d = bl[r * 4 + 2] + w[12] * r0 + w[13] * r1 + w[14] * r2 + w[15] * r3 + w[16] * r4 + w[17] * r5;
        l = fmaxf(d, 0.f); s2 += l * hwr; m2 += l;
        d = bl[r * 4 + 3] + w[18] * r0 + w[19] * r1 + w[20] * r2 + w[21] * r3 + w[22] * r4 + w[23] * r5;
        l = fmaxf(d, 0.f); s3 += l * hwr; m3 += l;
    }
    int ks = 0; float best = s0;
    if (s1 > best) { best = s1; ks = 1; }
    if (s2 > best) { best = s2; ks = 2; }
    if (s3 > best) { best = s3; ks = 3; }
    float sum = (ks == 0) ? m0 : (ks == 1) ? m1 : (ks == 2) ? m2 : m3;
    bool vis = visible[n * NN + i] > 0;
    sv[n * NN + i] = vis ? (sum + cs[ks]) : -1e30f;
    kstar[n * NN + i] = ks;
}

// ---------------------------------------------------------------------------
// K2: per n: per-head softmax over neighbors, social[n,:], then x = relu(...) + social
// grid NN, block 128 (thread = one r)
// ---------------------------------------------------------------------------
__global__ void k2_social_x(const float* __restrict__ nodes,
                            const float* __restrict__ W_loc,
                            const float* __restrict__ b_loc,
                            const float* __restrict__ W_mode,
                            const float* __restrict__ b_mode,
                            const float* __restrict__ W_in,
                            const float* __restrict__ b_in,
                            const float* __restrict__ sv,
                            const int* __restrict__ kstar,
                            float* __restrict__ xws) {
    __shared__ float wl[512 * 6];
    __shared__ float bl[512];
    __shared__ float wm[RR * KK];
    __shared__ float bm[RR];
    __shared__ float ndl[NN * 6];
    __shared__ float svl[NN];
    __shared__ int   kl[NN];
    __shared__ float wt[NN];
    __shared__ float mk[KK], dk[KK];
    int n = blockIdx.x, tid = threadIdx.x;
    for (int j = tid; j < 512 * 6; j += 128) wl[j] = W_loc[j];
    for (int j = tid; j < 512; j += 128) bl[j] = b_loc[j];
    for (int j = tid; j < RR * KK; j += 128) wm[j] = W_mode[j];
    bm[tid] = b_mode[tid];
    for (int j = tid; j < NN * 6; j += 128) ndl[j] = nodes[j];
    for (int i = tid; i < NN; i += 128) { svl[i] = sv[n * NN + i]; kl[i] = kstar[n * NN + i]; }
    __syncthreads();

    if (tid < KK) {
        float m = -3e38f;
        for (int i = 0; i < NN; ++i)
            if (kl[i] == tid && svl[i] > -1e29f) m = fmaxf(m, svl[i]);
        float d = 0.f;
        for (int i = 0; i < NN; ++i)
            if (kl[i] == tid && svl[i] > -1e29f) d += expf(svl[i] - m);
        mk[tid] = m;
        dk[tid] = (d > 0.f) ? d : 1.f;
    }
    __syncthreads();
    for (int i = tid; i < NN; i += 128) {
        float wv = 0.f;
        if (svl[i] > -1e29f) { int k = kl[i]; wv = expf(svl[i] - mk[k]) / dk[k]; }
        wt[i] = wv;
    }
    __syncthreads();

    int r = tid;
    float n0 = ndl[n * 6 + 0], n1 = ndl[n * 6 + 1], n2 = ndl[n * 6 + 2];
    float n3 = ndl[n * 6 + 3], n4 = ndl[n * 6 + 4], n5 = ndl[n * 6 + 5];
    float acc = 0.f;
    for (int i = 0; i < NN; ++i) {
        float wv = wt[i];
        if (wv != 0.f) {
            int k = kl[i];
            int j = r * 4 + k;
            const float* w = &wl[j * 6];
            float d = bl[j]
                + w[0] * (n0 - ndl[i * 6 + 0]) + w[1] * (n1 - ndl[i * 6 + 1])
                + w[2] * (n2 - ndl[i * 6 + 2]) + w[3] * (n3 - ndl[i * 6 + 3])
                + w[4] * (n4 - ndl[i * 6 + 4]) + w[5] * (n5 - ndl[i * 6 + 5]);
            float l = fmaxf(d, 0.f);
            acc += wv * (l + wm[r * 4 + k] + bm[r]);
        }
    }
    float pre = b_in[r] + n0 * W_in[r * 4 + 0] + n1 * W_in[r * 4 + 1]
                        + n2 * W_in[r * 4 + 2] + n3 * W_in[r * 4 + 3];
    xws[n * RR + r] = fmaxf(pre, 0.f) + acc;
}

// ---------------------------------------------------------------------------
// K3: gates = [x|h] (384x256) @ [W_ih|W_hh]^T (256x512) + b_ih + b_hh, WMMA f32
// grid (24, 4), block 256 = 8 waves; wave -> one 16x16 tile of one gate slab
// ---------------------------------------------------------------------------
__global__ void k3_gates(const float* __restrict__ xws,
                         const float* __restrict__ hin,
                         const float* __restrict__ W_ih,
                         const float* __restrict__ W_hh,
                         const float* __restrict__ b_ih,
                         const float* __restrict__ b_hh,
                         float* __restrict__ gates) {
    __shared__ float zt[16 * 256];
    int n0 = blockIdx.x * 16;
    int tid = threadIdx.x;
    for (int t = tid; t < 16 * 256; t += 256) {
        int m = t >> 8, tt = t & 255;
        zt[t] = (tt < 128) ? xws[(n0 + m) * RR + tt] : hin[(n0 + m) * RR + tt - 128];
    }
    __syncthreads();

    int wave = tid >> 5, lane = tid & 31;
    int col = lane & 15, half = lane >> 4;
    int j0 = blockIdx.y * 128 + wave * 16;

    v8f acc = {0.f, 0.f, 0.f, 0.f, 0.f, 0.f, 0.f, 0.f};
    for (int t0 = 0; t0 < 256; t0 += 4) {
        const float* Wsel = (t0 < 128) ? W_ih : W_hh;
        int tl = t0 & 127;
        v2f a, b;
        // A 16x4 f32 layout: lane = M, VGPR0 = K{0|2}, VGPR1 = K{1|3} per half
        a.x = zt[col * 256 + t0 + 2 * half];
        a.y = zt[col * 256 + t0 + 2 * half + 1];
        // B 4x16 f32 layout mirrors A: lane = N col
        b.x = Wsel[(j0 + col) * RR + tl + 2 * half];
        b.y = Wsel[(j0 + col) * RR + tl + 2 * half + 1];
        acc = __builtin_amdgcn_wmma_f32_16x16x4_f32(false, a, false, b,
                                                    (short)0, acc, false, false);
    }
    for (int v = 0; v < 8; ++v) {
        int row = n0 + v + 8 * half;
        int j = j0 + col;
        gates[row * 512 + j] = acc[v] + b_ih[j] + b_hh[j];
    }
}

// ---------------------------------------------------------------------------
// K4: LSTM pointwise: c = sig(f)*c_in + sig(i)*tanh(g); h = sig(o)*tanh(c)
// ---------------------------------------------------------------------------
__global__ void k4_lstm(const float* __restrict__ gates,
                        const float* __restrict__ cin,
                        float* __restrict__ hout,
                        float* __restrict__ cout) {
    int idx = blockIdx.x * blockDim.x + threadIdx.x;
    if (idx >= NN * RR) return;
    int n = idx >> 7, r = idx & 127;
    const float* g = &gates[n * 512];
    float ig = sigmoidf_(g[r]);
    float fg = sigmoidf_(g[128 + r]);
    float gg = tanhf(g[256 + r]);
    float og = sigmoidf_(g[384 + r]);
    float c = fg * cin[idx] + ig * gg;
    float h = og * tanhf(c);
    cout[idx] = c;
    hout[idx] = h;
}

// ---------------------------------------------------------------------------
// K5: head linear: h (384x128) @ [W_pi;W_mu;W_sig]^T (128x25, padded to 32), WMMA f32
// grid 24, block 64 = 2 waves (cols 0..15 / 16..31)
// ---------------------------------------------------------------------------
__global__ void k5_heads(const float* __restrict__ hsrc,
                         const float* __restrict__ W_pi,
                         const float* __restrict__ W_mu,
                         const float* __restrict__ W_sig,
                         float* __restrict__ headlin) {
    __shared__ float wh[32 * 128];
    __shared__ float ht[16 * 128];
    int n0 = blockIdx.x * 16;
    int tid = threadIdx.x;  // 64
    for (int t = tid; t < 32 * 128; t += 64) {
        int c = t >> 7, k = t & 127;
        float v = 0.f;
        if (c < 5)       v = W_pi[c * RR + k];
        else if (c < 15) v = W_mu[(c - 5) * RR + k];
        else if (c < 25) v = W_sig[(c - 15) * RR + k];
        wh[t] = v;
    }
    for (int t = tid; t < 16 * 128; t += 64) ht[t] = hsrc[n0 * RR + t];
    __syncthreads();

    int wave = tid >> 5, lane = tid & 31;
    int col = lane & 15, half = lane >> 4;
    int c0 = wave * 16;
    v8f acc = {0.f, 0.f, 0.f, 0.f, 0.f, 0.f, 0.f, 0.f};
    for (int t0 = 0; t0 < 128; t0 += 4) {
        v2f a, b;
        a.x = ht[col * 128 + t0 + 2 * half];
        a.y = ht[col * 128 + t0 + 2 * half + 1];
        b.x = wh[(c0 + col) * 128 + t0 + 2 * half];
        b.y = wh[(c0 + col) * 128 + t0 + 2 * half + 1];
        acc = __builtin_amdgcn_wmma_f32_16x16x4_f32(false, a, false, b,
                                                    (short)0, acc, false, false);
    }
    for (int v = 0; v < 8; ++v) {
        int row = n0 + v + 8 * half;
        headlin[row * 32 + c0 + col] = acc[v];
    }
}

// ---------------------------------------------------------------------------
// K6: pi = softmax(lin[0:5]+b_pi); mu = lin[5:15]+b_mu; sigma = exp(lin[15:25]+b_sig)
// ---------------------------------------------------------------------------
__global__ void k6_post(const float* __restrict__ headlin,
                        const float* __restrict__ b_pi,
                        const float* __restrict__ b_mu,
                        const float* __restrict__ b_sig,
                        float* __restrict__ pi,
                        float* __restrict__ mu,
                        float* __restrict__ sig) {
    int n = blockIdx.x, t = threadIdx.x;  // 32
    const float* lin = &headlin[n * 32];
    if (t < GG) {
        float m = -3e38f;
        for (int c = 0; c < GG; ++c) m = fmaxf(m, lin[c] + b_pi[c]);
        float den = 0.f;
        for (int c = 0; c < GG; ++c) den += expf(lin[c] + b_pi[c] - m);
        pi[n * GG + t] = expf(lin[t] + b_pi[t] - m) / den;
    } else if (t < 15) {
        int c = t - 5;
        mu[n * 10 + c] = lin[t] + b_mu[c];
    } else if (t < 25) {
        int c = t - 15;
        sig[n * 10 + c] = expf(lin[t] + b_sig[c]);
    }
}

// ---------------------------------------------------------------------------
extern "C" void kernel_launch(void* const* d_in, const int* in_sizes, int n_in,
                              void* d_out, int out_size, void* d_ws, size_t ws_size,
                              hipStream_t stream) {
    const float* nodes   = (const float*)d_in[0];
    const int*   visible = (const int*)  d_in[1];
    const float* hin     = (const float*)d_in[2];
    const float* cin     = (const float*)d_in[3];
    const float* W_loc   = (const float*)d_in[4];
    const float* b_loc   = (const float*)d_in[5];
    const float* W_score = (const float*)d_in[6];
    // d_in[7] = b_score: constant across K, cancels in argmax -> unused in forward
    const float* W_mode  = (const float*)d_in[8];
    const float* b_mode  = (const float*)d_in[9];
    const float* W_in    = (const float*)d_in[10];
    const float* b_in    = (const float*)d_in[11];
    const float* W_ih    = (const float*)d_in[12];
    const float* W_hh    = (const float*)d_in[13];
    const float* b_ih    = (const float*)d_in[14];
    const float* b_hh    = (const float*)d_in[15];
    const float* W_pi    = (const float*)d_in[16];
    const float* b_pi    = (const float*)d_in[17];
    const float* W_mu    = (const float*)d_in[18];
    const float* b_mu    = (const float*)d_in[19];
    const float* W_sig   = (const float*)d_in[20];
    const float* b_sig   = (const float*)d_in[21];

    float* out  = (float*)d_out;
    float* pi   = out;                 // 384*5
    float* mu   = out + 1920;          // 384*10
    float* sg   = out + 5760;          // 384*10
    float* hout = out + 9600;          // 384*128
    float* cout = out + 58752;         // 384*128

    float* wsf     = (float*)d_ws;
    float* csum    = wsf;                                   // 4 (pad 16)
    float* sv      = wsf + 16;                              // NN*NN
    int*   kstar   = (int*)(wsf + 16 + NN * NN);            // NN*NN
    float* xws     = wsf + 16 + 2 * NN * NN;                // NN*RR
    float* gates   = xws + NN * RR;                         // NN*512
    float* headlin = gates + NN * 512;                      // NN*32

    k0_csum   <<<1, 128, 0, stream>>>(W_mode, b_mode, csum);
    k1_pass1  <<<dim3(NN, 3), 128, 0, stream>>>(nodes, visible, hin, W_loc, b_loc,
                                                W_score, csum, sv, kstar);
    k2_social_x<<<NN, 128, 0, stream>>>(nodes, W_loc, b_loc, W_mode, b_mode,
                                        W_in, b_in, sv, kstar, xws);
    k3_gates  <<<dim3(24, 4), 256, 0, stream>>>(xws, hin, W_ih, W_hh, b_ih, b_hh, gates);
    k4_lstm   <<<192, 256, 0, stream>>>(gates, cin, hout, cout);
    k5_heads  <<<24, 64, 0, stream>>>(hout, W_pi, W_mu, W_sig, headlin);
    k6_post   <<<NN, 32, 0, stream>>>(headlin, b_pi, b_mu, b_sig, pi, mu, sg);
}